// ReduceMask_80470507258354
// MI455X (gfx1250) — compile-verified
//
#include <hip/hip_runtime.h>
#include <stdint.h>

// ---- static problem config (mirrors reference) ----
#define N_      32
#define H_      1024
#define W_      1024
#define BCH_    73
#define BCW_    73
#define TOTAL_  (N_ * BCH_ * BCW_)            // 170528
#define THRESH  0.9f
#define CHUNK_  1024
#define NCHUNK_ ((TOTAL_ + CHUNK_ - 1) / CHUNK_)  // 167

// ---------------------------------------------------------------------------
// Stage 1: per (n, block-row) workgroup.
// Async-DMA the 16 (clamped) image rows of the pooling window into LDS,
// separable max: column-max over rows, then 16-wide horizontal windowed max.
// ---------------------------------------------------------------------------
__global__ __launch_bounds__(256) void pool_flags_kernel(
    const float* __restrict__ mask, unsigned char* __restrict__ flags) {
  __shared__ float4 rows4[16 * 256];   // 64 KB staging; front 4 KB reused for colmax

  const int tid = threadIdx.x;
  const int i   = blockIdx.x;          // block row index (0..72)
  const int n   = blockIdx.y;          // batch index

  // Padded window rows map to real rows [i*14-1, i*14+14]; clamp to image.
  // Skipping pad rows is exact: data >= 0, so pad value 0 never raises the max.
  const int r0    = i * 14 - 1;
  const int rs    = (r0 < 0) ? 0 : r0;
  const int re    = (r0 + 15 > H_ - 1) ? (H_ - 1) : (r0 + 15);
  const int nrows = re - rs + 1;       // 15 or 16

  // Each lane owns a 16-byte (float4) column slice of every row.
  uint64_t gaddr = (uint64_t)(const void*)(mask + ((size_t)n * H_ + (size_t)rs) * W_)
                 + (uint64_t)((uint32_t)tid * 16u);
  uint32_t laddr = (uint32_t)(size_t)(&rows4[0]) + (uint32_t)tid * 16u;

  // CDNA5 async global->LDS DMA: up to 16 B128 transfers in flight per lane,
  // tracked by ASYNCcnt (no data VGPRs consumed while in flight).
  for (int k = 0; k < nrows; ++k) {
    asm volatile("global_load_async_to_lds_b128 %0, %1, off"
                 :: "v"(laddr + (uint32_t)k * 4096u),
                    "v"(gaddr + (uint64_t)k * (uint64_t)(W_ * 4))
                 : "memory");
  }
  asm volatile("s_wait_asynccnt 0x0" ::: "memory");

  // Column-wise max across the window rows (each lane reads only what it
  // loaded itself, so ASYNCcnt==0 is sufficient; no barrier needed yet).
  float4 a = make_float4(0.f, 0.f, 0.f, 0.f);
  for (int k = 0; k < nrows; ++k) {
    float4 v = rows4[k * 256 + tid];
    a.x = fmaxf(a.x, v.x);
    a.y = fmaxf(a.y, v.y);
    a.z = fmaxf(a.z, v.z);
    a.w = fmaxf(a.w, v.w);
  }

  // Overlay colmax[1024] into the front 4 KB of the staging buffer
  // (lane t overwrites only the slot lane t itself read).
  rows4[tid] = a;
  __syncthreads();

  if (tid < BCW_) {
    const float* colmax = (const float*)&rows4[0];
    // Padded window cols map to real cols [tid*14-1, tid*14+14]; clamp.
    int c0 = tid * 14 - 1;
    int cs = (c0 < 0) ? 0 : c0;
    int ce = (c0 + 15 > W_ - 1) ? (W_ - 1) : (c0 + 15);
    float m = 0.f;
    for (int c = cs; c <= ce; ++c) m = fmaxf(m, colmax[c]);
    flags[((size_t)n * BCH_ + (size_t)i) * BCW_ + (size_t)tid] =
        (m > THRESH) ? (unsigned char)1 : (unsigned char)0;
  }
}

// ---------------------------------------------------------------------------
// Stage 2: output is padded with -1 (must be rewritten every call).
// ---------------------------------------------------------------------------
__global__ void fill_neg1_kernel(int* __restrict__ out, int count) {
  int idx = blockIdx.x * blockDim.x + threadIdx.x;
  if (idx < count) out[idx] = -1;
}

// ---------------------------------------------------------------------------
// Stage 3a: per-chunk population counts (167 WGs, wave32 ballot + WG reduce).
// ---------------------------------------------------------------------------
__global__ __launch_bounds__(1024) void chunk_count_kernel(
    const unsigned char* __restrict__ flags, int* __restrict__ chunkSum) {
  __shared__ int wsum[32];
  const int tid = threadIdx.x;
  const int c   = blockIdx.x;
  const int p   = c * CHUNK_ + tid;
  const int f   = (p < TOTAL_) ? (int)flags[p] : 0;

  unsigned bal = __builtin_amdgcn_ballot_w32(f != 0);
  if ((tid & 31) == 0) wsum[tid >> 5] = __popc(bal);
  __syncthreads();

  if (tid == 0) {
    int s = 0;
#pragma unroll
    for (int w = 0; w < 32; ++w) s += wsum[w];
    chunkSum[c] = s;
  }
}

// ---------------------------------------------------------------------------
// Stage 3b: exclusive scan over the 167 chunk counts (tiny).
// ---------------------------------------------------------------------------
__global__ void scan_chunks_kernel(const int* __restrict__ chunkSum,
                                   int* __restrict__ chunkBase) {
  if (threadIdx.x == 0 && blockIdx.x == 0) {
    int run = 0;
    for (int c = 0; c < NCHUNK_; ++c) {
      chunkBase[c] = run;
      run += chunkSum[c];
    }
  }
}

// ---------------------------------------------------------------------------
// Stage 3c: ordered scatter. Rank = chunkBase + wave offset + lane prefix,
// exactly reproducing jnp.nonzero's row-major order.
// ---------------------------------------------------------------------------
__global__ __launch_bounds__(1024) void scatter_kernel(
    const unsigned char* __restrict__ flags, const int* __restrict__ chunkBase,
    int* __restrict__ out) {
  __shared__ int wsum[32];
  __shared__ int woff[32];
  const int tid  = threadIdx.x;
  const int lane = tid & 31;
  const int wid  = tid >> 5;
  const int c    = blockIdx.x;
  const int p    = c * CHUNK_ + tid;
  const int f    = (p < TOTAL_) ? (int)flags[p] : 0;

  unsigned bal = __builtin_amdgcn_ballot_w32(f != 0);
  int pre = __popc(bal & ((1u << lane) - 1u));
  if (lane == 0) wsum[wid] = __popc(bal);
  __syncthreads();

  if (tid == 0) {
    int run = chunkBase[c];
#pragma unroll
    for (int w = 0; w < 32; ++w) { woff[w] = run; run += wsum[w]; }
  }
  __syncthreads();

  if (f) {
    int rank = woff[wid] + pre;
    int nn   = p / (BCH_ * BCW_);
    int rem  = p - nn * (BCH_ * BCW_);
    int bh   = rem / BCW_;
    int bw   = rem - bh * BCW_;
    out[3 * rank + 0] = nn;
    out[3 * rank + 1] = bh;
    out[3 * rank + 2] = bw;
  }
}

// ---------------------------------------------------------------------------
extern "C" void kernel_launch(void* const* d_in, const int* in_sizes, int n_in,
                              void* d_out, int out_size, void* d_ws, size_t ws_size,
                              hipStream_t stream) {
  (void)in_sizes; (void)n_in; (void)out_size; (void)ws_size;

  const float* mask = (const float*)d_in[0];
  int* out          = (int*)d_out;

  // Workspace layout (all within d_ws):
  //   [0, 170528)        : flags, 1 byte per pooled block
  //   [171008, +167*4)   : chunkSum
  //   [...,    +167*4)   : chunkBase
  unsigned char* flags = (unsigned char*)d_ws;
  int* chunkSum  = (int*)((char*)d_ws + (size_t)NCHUNK_ * CHUNK_);  // 171008, 4B aligned
  int* chunkBase = chunkSum + NCHUNK_;

  // 1) pooled-threshold flags (bandwidth stage, CDNA5 async-DMA path)
  pool_flags_kernel<<<dim3(BCH_, N_), 256, 0, stream>>>(mask, flags);

  // 2) pad output with -1
  const int cnt = TOTAL_ * 3;
  fill_neg1_kernel<<<(cnt + 255) / 256, 256, 0, stream>>>(out, cnt);

  // 3) ordered compaction: count -> scan -> scatter (parallel count/scatter, single-thread scan)
  chunk_count_kernel<<<NCHUNK_, CHUNK_, 0, stream>>>(flags, chunkSum);
  scan_chunks_kernel<<<1, 32, 0, stream>>>(chunkSum, chunkBase);
  scatter_kernel<<<NCHUNK_, CHUNK_, 0, stream>>>(flags, chunkBase, out);
}